// PGA_Unfold_J10_RMSProp_41772851921187
// MI455X (gfx1250) — compile-verified
//
#include <hip/hip_runtime.h>
#include <math.h>

// Problem sizes (A=1, B from grid)
#define MU 8     // users
#define NN 64    // antennas
#define RR 16    // RF chains

// Constants from the reference
#define LN2f     0.69314718056f
#define SIGMA2f  1.0f
#define BETAf    0.9f
#define EPSf     1e-8f
#define ETA_Ff   0.01f
#define ETA_Wf   0.01f
#define W_F_COMf 1.0f
#define W_F_CRBf 0.5f
#define W_W_COMf 1.0f
#define W_W_CRBf 0.5f

typedef float v2f __attribute__((ext_vector_type(2)));
typedef float v8f __attribute__((ext_vector_type(8)));

// ---------------- complex helpers ----------------
struct c2 { float re, im; };
__device__ __forceinline__ c2 cadd(c2 a, c2 b) { return {a.re + b.re, a.im + b.im}; }
__device__ __forceinline__ c2 cmul(c2 a, c2 b) { return {a.re*b.re - a.im*b.im, a.re*b.im + a.im*b.re}; }
__device__ __forceinline__ c2 csqrt_(c2 z) {
  float m = sqrtf(z.re*z.re + z.im*z.im);
  float x = sqrtf(fmaxf(0.5f*(m + z.re), 0.0f));
  float y = copysignf(sqrtf(fmaxf(0.5f*(m - z.re), 0.0f)), z.im);
  return {x, y};
}
__device__ __forceinline__ c2 cdiv_(c2 a, c2 b) {
  float inv = 1.0f / (b.re*b.re + b.im*b.im + 1e-30f);
  return {(a.re*b.re + a.im*b.im)*inv, (a.im*b.re - a.re*b.im)*inv};
}

// wave32 butterfly reduction (result broadcast to all lanes)
__device__ __forceinline__ float wred(float v) {
#pragma unroll
  for (int off = 16; off > 0; off >>= 1) v += __shfl_xor(v, off, 32);
  return v;
}

// ---------------- WMMA f32 16x16x4 complex tile machinery ----------------
struct CTile { v8f re, im; };
__device__ __forceinline__ void ctz(CTile& c) {
#pragma unroll
  for (int i = 0; i < 8; ++i) { c.re[i] = 0.0f; c.im[i] = 0.0f; }
}

__device__ __forceinline__ v8f wmma4(v2f a, v2f b, v8f c) {
  // D = A * B + C    (16x16x4, f32). NEG_A/NEG_B must be 0 for f32 per ISA.
  return __builtin_amdgcn_wmma_f32_16x16x4_f32(false, a, false, b, (short)0, c, false, false);
}

// Complex MMA: C += op(A)*op(B); conjugation/subtraction applied as VALU
// negations on the tiny per-lane fragments (f32 WMMA has no A/B NEG modifier).
template<bool CONJA, bool CONJB>
__device__ __forceinline__ void cmma4(CTile& C, v2f ar, v2f ai, v2f br, v2f bi) {
  v2f aim = CONJA ? -ai : ai;
  v2f bim = CONJB ? -bi : bi;
  C.re = wmma4(ar, br, C.re);
  C.re = wmma4(-aim, bim, C.re);   // re -= aim*bim
  C.im = wmma4(ar, bim, C.im);
  C.im = wmma4(aim, br, C.im);
}

// A fragment (16x4) at (row0,k0) from row-major src with leading dim ld.
// ISA layout: lanes 0-15 -> rows, VGPR halves -> K pairs {k0,k0+1} / {k0+2,k0+3}.
__device__ __forceinline__ v2f lda_frag(const float* src, int ld, int row0, int k0, int lane) {
  int r = lane & 15, h = lane >> 4;
  v2f a;
  a.x = src[(row0 + r)*ld + k0 + 2*h];
  a.y = src[(row0 + r)*ld + k0 + 2*h + 1];
  return a;
}
// A fragment from a transposed source: logical A[i][k] = src[k*ld + i]
__device__ __forceinline__ v2f lda_fragT(const float* src, int ld, int row0, int k0, int lane) {
  int r = lane & 15, h = lane >> 4;
  v2f a;
  a.x = src[(k0 + 2*h)*ld + row0 + r];
  a.y = src[(k0 + 2*h + 1)*ld + row0 + r];
  return a;
}
// B fragment (4x16) at (k0,col0), row-major src.
__device__ __forceinline__ v2f ldb_frag(const float* src, int ld, int k0, int col0, int lane) {
  int c = lane & 15, h = lane >> 4;
  v2f b;
  b.x = src[(k0 + 2*h)*ld + col0 + c];
  b.y = src[(k0 + 2*h + 1)*ld + col0 + c];
  return b;
}
// C/D layout: VGPR v holds rows v (lanes 0-15) and v+8 (lanes 16-31).
__device__ __forceinline__ void store_ctile(float* dre, float* dim_, int ld, int row0, int col0,
                                            const CTile& C, int lane) {
  int c = lane & 15, h = lane >> 4;
#pragma unroll
  for (int v = 0; v < 8; ++v) {
    int r = row0 + h*8 + v;
    dre[r*ld + col0 + c] = C.re[v];
    dim_[r*ld + col0 + c] = C.im[v];
  }
}
__device__ __forceinline__ void acc_ctile(float* dre, float* dim_, int ld, int row0, int col0,
                                          const CTile& C, float s, int lane) {
  int c = lane & 15, h = lane >> 4;
#pragma unroll
  for (int v = 0; v < 8; ++v) {
    int r = row0 + h*8 + v;
    dre[r*ld + col0 + c] += s * C.re[v];
    dim_[r*ld + col0 + c] += s * C.im[v];
  }
}

// ---------------- kernel 1: Mm = A_dot^H R_N_inv A_dot (batch-invariant) ----------------
__global__ void __launch_bounds__(256)
mm_precompute_kernel(const float* __restrict__ Ar, const float* __restrict__ Ai,
                     const float* __restrict__ Rr, const float* __restrict__ Ri,
                     float* __restrict__ Mr, float* __restrict__ Mi) {
  __shared__ float Tr[64*64];
  __shared__ float Ti[64*64];
  int tid = threadIdx.x;
  // T = R_N_inv @ A_dot
  for (int idx = tid; idx < 64*64; idx += 256) {
    int l = idx >> 6, n = idx & 63;
    float accr = 0.f, acci = 0.f;
    for (int k = 0; k < 64; ++k) {
      float rr = Rr[l*64 + k], ri = Ri[l*64 + k];
      float br = Ar[k*64 + n], bi = Ai[k*64 + n];
      accr += rr*br - ri*bi;
      acci += rr*bi + ri*br;
    }
    Tr[idx] = accr; Ti[idx] = acci;
  }
  __syncthreads();
  // Mm = A_dot^H @ T
  for (int idx = tid; idx < 64*64; idx += 256) {
    int n = idx >> 6, n2 = idx & 63;
    float accr = 0.f, acci = 0.f;
    for (int l = 0; l < 64; ++l) {
      float ar = Ar[l*64 + n], ai = Ai[l*64 + n];
      float tr = Tr[l*64 + n2], ti = Ti[l*64 + n2];
      accr += ar*tr + ai*ti;
      acci += ar*ti - ai*tr;
    }
    Mr[idx] = accr; Mi[idx] = acci;
  }
}

// ---------------- kernel 2: one wave32 per batch element ----------------
__global__ void __launch_bounds__(32)
pga_unfold_kernel(const float* __restrict__ Hre_g, const float* __restrict__ Him_g,
                  const float* __restrict__ F0re_g, const float* __restrict__ F0im_g,
                  const float* __restrict__ W0re_g, const float* __restrict__ W0im_g,
                  const float* __restrict__ MmR, const float* __restrict__ MmI,
                  const float* __restrict__ xi0_g,
                  const int* __restrict__ Pt_g,
                  const int* __restrict__ no_g,
                  const int* __restrict__ ni_g,
                  float* __restrict__ out) {
  __shared__ float Fre[NN*RR], Fim[NN*RR];
  __shared__ float sFre[NN*RR], sFim[NN*RR];
  __shared__ float gFre[NN*RR], gFim[NN*RR];
  __shared__ float MmFre[NN*RR], MmFim[NN*RR];
  __shared__ float Hre[MU*NN], Him[MU*NN];
  __shared__ float Wre[RR*MU], Wim[RR*MU];
  __shared__ float sWre[RR*MU], sWim[RR*MU];
  __shared__ float gWre[RR*MU], gWim[RR*MU];
  __shared__ float Vre[RR*RR], Vim[RR*RR];
  __shared__ float Xre[RR*RR], Xim[RR*RR];   // F^H Mm F
  __shared__ float hHFre[MU*RR], hHFim[MU*RR];
  __shared__ float coefre[RR], coefim[RR];
  __shared__ float Gmat[MU*MU];
  __shared__ float rate_buf[32], crb_buf[32];

  const int lane = threadIdx.x;
  const int b    = blockIdx.x;
  const int Bn   = gridDim.x;

  const float* Hb_re = Hre_g  + (size_t)b * MU * NN;
  const float* Hb_im = Him_g  + (size_t)b * MU * NN;
  const float* Fb_re = F0re_g + (size_t)b * NN * RR;
  const float* Fb_im = F0im_g + (size_t)b * NN * RR;
  const float* Wb_re = W0re_g + (size_t)b * RR * MU;
  const float* Wb_im = W0im_g + (size_t)b * RR * MU;

  __builtin_prefetch(Hb_re, 0, 3);
  __builtin_prefetch(Hb_im, 0, 3);
  __builtin_prefetch(Fb_re, 0, 3);
  __builtin_prefetch(Fb_im, 0, 3);
  __builtin_prefetch(MmR, 0, 3);
  __builtin_prefetch(MmI, 0, 3);

  // Load batch state to LDS; zero momentum.
  for (int idx = lane; idx < NN*RR; idx += 32) {
    Fre[idx] = Fb_re[idx]; Fim[idx] = Fb_im[idx];
    sFre[idx] = 0.f; sFim[idx] = 0.f;
  }
  for (int idx = lane; idx < MU*NN; idx += 32) { Hre[idx] = Hb_re[idx]; Him[idx] = Hb_im[idx]; }
  for (int idx = lane; idx < RR*MU; idx += 32) {
    Wre[idx] = Wb_re[idx]; Wim[idx] = Wb_im[idx];
    sWre[idx] = 0.f; sWim[idx] = 0.f;
  }
  rate_buf[lane] = 0.f; crb_buf[lane] = 0.f;
  __syncthreads();

  const float Ptf = (float)Pt_g[0];
  const int   NO  = no_g[0];
  const int   NI  = ni_g[0];
  const float xi  = xi0_g[0];
  float trv = 1.0f;  // Re(trace(W^H F^H Mm F W)) for current state

  auto projectF = [&]() {
    for (int idx = lane; idx < NN*RR; idx += 32) {
      float r = Fre[idx], i = Fim[idx];
      float mag = sqrtf(r*r + i*i);
      if (mag > 1e-12f) { Fre[idx] = r/mag; Fim[idx] = i/mag; }
      else              { Fre[idx] = 0.f;   Fim[idx] = 0.f;  }
    }
    __syncthreads();
  };

  auto powerFW = [&]() -> float {   // ||F W||_F^2
    float p = 0.f;
    for (int idx = lane; idx < NN*MU; idx += 32) {
      int n = idx >> 3, m = idx & 7;
      float ar = 0.f, ai = 0.f;
      for (int r = 0; r < RR; ++r) {
        float fr = Fre[n*RR + r], fi = Fim[n*RR + r];
        float wr = Wre[r*MU + m], wi = Wim[r*MU + m];
        ar += fr*wr - fi*wi;
        ai += fr*wi + fi*wr;
      }
      p += ar*ar + ai*ai;
    }
    return wred(p);
  };

  auto computeV = [&]() {           // V = W W^H
    for (int idx = lane; idx < RR*RR; idx += 32) {
      int i = idx >> 4, j = idx & 15;
      float ar = 0.f, ai = 0.f;
      for (int m = 0; m < MU; ++m) {
        float a_r = Wre[i*MU + m], a_i = Wim[i*MU + m];
        float b_r = Wre[j*MU + m], b_i = Wim[j*MU + m];
        ar += a_r*b_r + a_i*b_i;
        ai += a_i*b_r - a_r*b_i;
      }
      Vre[idx] = ar; Vim[idx] = ai;
    }
    __syncthreads();
  };

  auto computehHF = [&]() {         // hHF[m][r] = sum_n conj(H[m][n]) F[n][r]
    for (int idx = lane; idx < MU*RR; idx += 32) {
      int m = idx >> 4, r = idx & 15;
      float ar = 0.f, ai = 0.f;
      for (int n = 0; n < NN; ++n) {
        float hr = Hre[m*NN + n], hi = Him[m*NN + n];
        float fr = Fre[n*RR + r], fi = Fim[n*RR + r];
        ar += hr*fr + hi*fi;
        ai += hr*fi - hi*fr;
      }
      hHFre[idx] = ar; hHFim[idx] = ai;
    }
    __syncthreads();
  };

  auto wmma_MmF = [&]() {           // MmF = Mm(64x64, global/L2) @ F(64x16)
    for (int mt = 0; mt < 4; ++mt) {
      CTile C; ctz(C);
      for (int k0 = 0; k0 < NN; k0 += 4) {
        v2f ar = lda_frag(MmR, NN, mt*16, k0, lane);
        v2f ai = lda_frag(MmI, NN, mt*16, k0, lane);
        v2f br = ldb_frag(Fre, RR, k0, 0, lane);
        v2f bi = ldb_frag(Fim, RR, k0, 0, lane);
        cmma4<false, false>(C, ar, ai, br, bi);
      }
      store_ctile(MmFre, MmFim, RR, mt*16, 0, C, lane);
    }
    __syncthreads();
  };

  auto wmma_FHMF = [&]() {          // X = F^H @ MmF  (16x16)
    CTile C; ctz(C);
    for (int k0 = 0; k0 < NN; k0 += 4) {
      v2f ar = lda_fragT(Fre, RR, 0, k0, lane);
      v2f ai = lda_fragT(Fim, RR, 0, k0, lane);
      v2f br = ldb_frag(MmFre, RR, k0, 0, lane);
      v2f bi = ldb_frag(MmFim, RR, k0, 0, lane);
      cmma4<true, false>(C, ar, ai, br, bi);
    }
    store_ctile(Xre, Xim, RR, 0, 0, C, lane);
    __syncthreads();
  };

  auto traceWXW = [&]() -> float {  // Re(trace(W^H X W))
    float tp = 0.f;
    for (int idx = lane; idx < RR*MU; idx += 32) {
      int i = idx >> 3, m = idx & 7;
      float ar = 0.f, ai = 0.f;
      for (int j = 0; j < RR; ++j) {
        float xr = Xre[i*RR + j], xj = Xim[i*RR + j];
        float wr = Wre[j*MU + m], wi = Wim[j*MU + m];
        ar += xr*wr - xj*wi;
        ai += xr*wi + xj*wr;
      }
      tp += Wre[i*MU + m]*ar + Wim[i*MU + m]*ai;
    }
    return wred(tp);
  };

  auto trackRate = [&](int slot) {  // sum-rate + CRB for current (hHF, W, trv)
    for (int idx = lane; idx < MU*MU; idx += 32) {
      int m = idx >> 3, k = idx & 7;
      float ar = 0.f, ai = 0.f;
      for (int r = 0; r < RR; ++r) {
        float hr = hHFre[m*RR + r], hi = hHFim[m*RR + r];
        float wr = Wre[r*MU + k],  wi = Wim[r*MU + k];
        ar += hr*wr - hi*wi;
        ai += hr*wi + hi*wr;
      }
      Gmat[idx] = ar*ar + ai*ai;
    }
    __syncthreads();
    float rs = 0.f;
    if (lane < MU) {
      int m = lane;
      float sig = Gmat[m*MU + m];
      float tot = 0.f;
      for (int k = 0; k < MU; ++k) tot += Gmat[m*MU + k];
      rs = log2f(1.0f + sig / (tot - sig + SIGMA2f));
    }
    rs = wred(rs);
    if (lane == 0) {
      rate_buf[slot] = rs;
      crb_buf[slot]  = 1.0f / (2.0f * xi * xi * trv + 1e-12f);
    }
    __syncthreads();
  };

  // ---- init: project F, normalize power (scale W) ----
  projectF();
  {
    float p = powerFW();
    float sc = sqrtf(Ptf / p);
    for (int idx = lane; idx < RR*MU; idx += 32) { Wre[idx] *= sc; Wim[idx] *= sc; }
    __syncthreads();
  }

  for (int ii = 0; ii < NO; ++ii) {
    computeV();
    computehHF();
    wmma_MmF(); wmma_FHMF(); trv = traceWXW();

    for (int jj = 0; jj < NI; ++jj) {
      for (int idx = lane; idx < NN*RR; idx += 32) { gFre[idx] = 0.f; gFim[idx] = 0.f; }
      __syncthreads();

      // ---- grad_F_com: rank-1 structure -> per-user outer products ----
      for (int m = 0; m < MU; ++m) {
        float sre_p = 0.f, sim_p = 0.f;
        if (lane < RR) {
          int j = lane;
          c2 hf = {hHFre[m*RR + j], hHFim[m*RR + j]};
          c2 wj = {Wre[j*MU + m],  Wim[j*MU + m]};
          c2 t  = cmul(hf, wj);
          sre_p = t.re; sim_p = t.im;
        }
        float sre = wred(sre_p), sim = wred(sim_p);  // s = hHF . w_m
        c2 u = {0.f, 0.f}, u2 = {0.f, 0.f};
        float q1p = 0.f, q2p = 0.f;
        if (lane < RR) {
          int r = lane;
          for (int j = 0; j < RR; ++j) {
            c2 hf = {hHFre[m*RR + j], hHFim[m*RR + j]};
            c2 v  = {Vre[j*RR + r],  Vim[j*RR + r]};
            u = cadd(u, cmul(hf, v));                // u = hHF V
          }
          c2 wc = {Wre[r*MU + m], -Wim[r*MU + m]};
          c2 sw = cmul({sre, sim}, wc);
          u2 = {u.re - sw.re, u.im - sw.im};         // u2 = hHF V_mk
          c2 hf = {hHFre[m*RR + r], hHFim[m*RR + r]};
          q1p = u.re*hf.re + u.im*hf.im;             // Re(u . conj(hHF))
          q2p = u2.re*hf.re + u2.im*hf.im;
        }
        float q1 = wred(q1p), q2 = wred(q2p);
        float d1 = LN2f*(q1 + SIGMA2f) + 1e-4f;
        float d2 = LN2f*(q2 + SIGMA2f) + 1e-4f;
        if (lane < RR) {
          coefre[lane] = u.re/d1 - u2.re/d2;
          coefim[lane] = u.im/d1 - u2.im/d2;
        }
        __syncthreads();
        for (int idx = lane; idx < NN*RR; idx += 32) {
          int n = idx >> 4, r = idx & 15;
          c2 h  = {Hre[m*NN + n], Him[m*NN + n]};
          c2 cf = {coefre[r], coefim[r]};
          c2 g  = cmul(h, cf);                       // h (x) coef
          gFre[idx] += W_F_COMf * g.re;
          gFim[idx] += W_F_COMf * g.im;
        }
        __syncthreads();
      }

      // ---- grad_F_crb: gF += (0.5/tr) * MmF @ V  (WMMA) ----
      {
        float sc = W_F_CRBf / trv;
        for (int mt = 0; mt < 4; ++mt) {
          CTile C; ctz(C);
          for (int k0 = 0; k0 < RR; k0 += 4) {
            v2f ar = lda_frag(MmFre, RR, mt*16, k0, lane);
            v2f ai = lda_frag(MmFim, RR, mt*16, k0, lane);
            v2f br = ldb_frag(Vre, RR, k0, 0, lane);
            v2f bi = ldb_frag(Vim, RR, k0, 0, lane);
            cmma4<false, false>(C, ar, ai, br, bi);
          }
          acc_ctile(gFre, gFim, RR, mt*16, 0, C, sc, lane);
        }
        __syncthreads();
      }

      // ---- RMSProp F update (complex sqrt per reference) ----
      for (int idx = lane; idx < NN*RR; idx += 32) {
        c2 g = {gFre[idx], gFim[idx]};
        c2 s = {BETAf*sFre[idx] + (1.f - BETAf)*g.re,
                BETAf*sFim[idx] + (1.f - BETAf)*g.im};
        sFre[idx] = s.re; sFim[idx] = s.im;
        c2 sq = csqrt_(s); sq.re += EPSf;
        c2 upd = cdiv_(g, sq);
        Fre[idx] += ETA_Ff * upd.re;
        Fim[idx] += ETA_Ff * upd.im;
      }
      __syncthreads();

      // ---- normalize_power (scale F) ----
      {
        float p = powerFW();
        float sc = sqrtf(Ptf / p);
        for (int idx = lane; idx < NN*RR; idx += 32) { Fre[idx] *= sc; Fim[idx] *= sc; }
        __syncthreads();
      }

      // ---- refresh state for tracking AND next iteration ----
      computehHF();
      wmma_MmF(); wmma_FHMF(); trv = traceWXW();
      trackRate(ii*(NI + 1) + jj + 1);
    }

    // ---- outer: W update with projected F ----
    projectF();
    computehHF();
    wmma_MmF(); wmma_FHMF(); trv = traceWXW();

    for (int idx = lane; idx < RR*MU; idx += 32) { gWre[idx] = 0.f; gWim[idx] = 0.f; }
    __syncthreads();

    // grad_W_crb = (X @ W) / tr
    {
      float sc = W_W_CRBf / trv;
      for (int idx = lane; idx < RR*MU; idx += 32) {
        int i = idx >> 3, m = idx & 7;
        float ar = 0.f, ai = 0.f;
        for (int j = 0; j < RR; ++j) {
          float xr = Xre[i*RR + j], xj = Xim[i*RR + j];
          float wr = Wre[j*MU + m], wi = Wim[j*MU + m];
          ar += xr*wr - xj*wi;
          ai += xr*wi + xj*wr;
        }
        gWre[idx] += sc * ar;
        gWim[idx] += sc * ai;
      }
      __syncthreads();
    }

    // grad_W_com: rank-1, column m only
    for (int m = 0; m < MU; ++m) {
      float sre_p = 0.f, sim_p = 0.f, q1p = 0.f;
      if (lane < RR) {
        int j = lane;
        c2 hf = {hHFre[m*RR + j], hHFim[m*RR + j]};
        c2 wj = {Wre[j*MU + m],  Wim[j*MU + m]};
        c2 t  = cmul(hf, wj);
        sre_p = t.re; sim_p = t.im;
        c2 u = {0.f, 0.f};
        for (int k = 0; k < RR; ++k) {
          c2 hk = {hHFre[m*RR + k], hHFim[m*RR + k]};
          c2 v  = {Vre[k*RR + j],  Vim[k*RR + j]};
          u = cadd(u, cmul(hk, v));
        }
        q1p = u.re*hf.re + u.im*hf.im;   // trace(V Hbar_m)
      }
      float sre = wred(sre_p), sim = wred(sim_p);
      float q1  = wred(q1p);
      float denom = LN2f*(q1 + SIGMA2f);
      if (lane < RR) {
        int i = lane;
        c2 fh = {hHFre[m*RR + i], -hHFim[m*RR + i]};  // F^H h = conj(hHF)
        c2 contrib = cmul(fh, {sre, sim});
        float w = W_W_COMf / (denom * (float)MU);
        gWre[i*MU + m] += w * contrib.re;
        gWim[i*MU + m] += w * contrib.im;
      }
      __syncthreads();
    }

    // RMSProp W update
    for (int idx = lane; idx < RR*MU; idx += 32) {
      c2 g = {gWre[idx], gWim[idx]};
      c2 s = {BETAf*sWre[idx] + (1.f - BETAf)*g.re,
              BETAf*sWim[idx] + (1.f - BETAf)*g.im};
      sWre[idx] = s.re; sWim[idx] = s.im;
      c2 sq = csqrt_(s); sq.re += EPSf;
      c2 upd = cdiv_(g, sq);
      Wre[idx] += ETA_Wf * upd.re;
      Wim[idx] += ETA_Wf * upd.im;
    }
    __syncthreads();

    // normalize (scale W)
    {
      float p = powerFW();
      float sc = sqrtf(Ptf / p);
      for (int idx = lane; idx < RR*MU; idx += 32) { Wre[idx] *= sc; Wim[idx] *= sc; }
      __syncthreads();
    }

    // X depends on F only -> just refresh trace with the new W, then track.
    trv = traceWXW();
    trackRate(ii*(NI + 1));
  }

  // ---- outputs: rates[B,T], crbs[B,T], F (interleaved c64), W (interleaved c64) ----
  int T = NO*(NI + 1);
  for (int t = lane; t < T; t += 32) {
    out[(size_t)b*T + t]                    = rate_buf[t];
    out[(size_t)Bn*T + (size_t)b*T + t]     = crb_buf[t];
  }
  size_t fbase = 2ull * (size_t)Bn * (size_t)T;
  for (int idx = lane; idx < NN*RR; idx += 32) {
    out[fbase + ((size_t)b*NN*RR + idx)*2 + 0] = Fre[idx];
    out[fbase + ((size_t)b*NN*RR + idx)*2 + 1] = Fim[idx];
  }
  size_t wbase = fbase + 2ull * (size_t)Bn * (size_t)(NN*RR);
  for (int idx = lane; idx < RR*MU; idx += 32) {
    out[wbase + ((size_t)b*RR*MU + idx)*2 + 0] = Wre[idx];
    out[wbase + ((size_t)b*RR*MU + idx)*2 + 1] = Wim[idx];
  }
}

extern "C" void kernel_launch(void* const* d_in, const int* in_sizes, int n_in,
                              void* d_out, int out_size, void* d_ws, size_t ws_size,
                              hipStream_t stream) {
  (void)n_in; (void)out_size; (void)ws_size;
  const float* Hre  = (const float*)d_in[0];
  const float* Him  = (const float*)d_in[1];
  const float* F0re = (const float*)d_in[2];
  const float* F0im = (const float*)d_in[3];
  const float* W0re = (const float*)d_in[4];
  const float* W0im = (const float*)d_in[5];
  const float* Ar   = (const float*)d_in[6];
  const float* Ai   = (const float*)d_in[7];
  const float* Rr   = (const float*)d_in[8];
  const float* Ri   = (const float*)d_in[9];
  const float* xi0  = (const float*)d_in[10];
  const int*   Pt   = (const int*)d_in[11];
  const int*   no   = (const int*)d_in[12];
  const int*   ni   = (const int*)d_in[13];

  float* Mr = (float*)d_ws;          // 64x64 re
  float* Mi = Mr + 64*64;            // 64x64 im

  mm_precompute_kernel<<<1, 256, 0, stream>>>(Ar, Ai, Rr, Ri, Mr, Mi);

  int B = in_sizes[0] / (MU*NN);     // A*B = 1024 batch problems
  pga_unfold_kernel<<<B, 32, 0, stream>>>(Hre, Him, F0re, F0im, W0re, W0im,
                                          Mr, Mi, xi0, Pt, no, ni, (float*)d_out);
}